// GUP_4105988735544
// MI455X (gfx1250) — compile-verified
//
#include <hip/hip_runtime.h>
#include <math.h>

// Problem dims (fixed by the reference)
#define Bb 32
#define Mm 6
#define AQ 128
#define LK 512
#define Dd 128
#define Hh 8
#define HD 16

typedef __attribute__((ext_vector_type(2))) float v2f;
typedef __attribute__((ext_vector_type(8))) float v8f;

// D(16x16,f32) = A(16x4,f32) * B(4x16,f32) + C  — wave32 CDNA5 WMMA
// Fragment layout (ISA 7.12.2): lane lo=lane&15, hi=lane>>4
//   A vgpr j : A[lo][2*hi + j]      (k index)
//   B vgpr j : B[2*hi + j][lo]      (k x n)
//   D vgpr r : D[r + 8*hi][lo]
static __device__ __forceinline__ v8f wmma4(v2f a, v2f b, v8f c) {
  return __builtin_amdgcn_wmma_f32_16x16x4_f32(
      /*neg_a=*/false, a, /*neg_b=*/false, b,
      /*c_mod=*/(short)0, c, /*reuse_a=*/false, /*reuse_b=*/false);
}

// ---------------------------------------------------------------------------
// Y[R x 128] = X[R x 128] @ W^T (+ bias).  W is [128 x 128] row-major (n,k).
// Block: 256 threads = 8 waves. Block handles 16 rows; wave w -> cols 16w..16w+15.
// K loop: 32 x wmma_f32_16x16x4.
// ---------------------------------------------------------------------------
__global__ void __launch_bounds__(256) proj128(const float* __restrict__ X,
                                               const float* __restrict__ W,
                                               const float* __restrict__ bias,
                                               float* __restrict__ Y) {
  const int lane = threadIdx.x & 31;
  const int wave = threadIdx.x >> 5;
  const int lo = lane & 15, hi = lane >> 4;
  const int row_base = blockIdx.x * 16;
  const int col_base = wave * 16;

  const float* xr = X + (size_t)(row_base + lo) * Dd; // A row for this lane
  const float* wr = W + (size_t)(col_base + lo) * Dd; // B^T row (W[n][k])

  v8f acc = {0.f, 0.f, 0.f, 0.f, 0.f, 0.f, 0.f, 0.f};
#pragma unroll 4
  for (int kk = 0; kk < Dd / 4; ++kk) {
    const int k0 = 4 * kk + 2 * hi;
    v2f a = *(const v2f*)(xr + k0);
    v2f b = *(const v2f*)(wr + k0);
    acc = wmma4(a, b, acc);
  }
  const float bn = bias ? bias[col_base + lo] : 0.0f;
#pragma unroll
  for (int r = 0; r < 8; ++r) {
    const int m = r + 8 * hi;
    Y[(size_t)(row_base + m) * Dd + col_base + lo] = acc[r] + bn;
  }
}

// ---------------------------------------------------------------------------
// Fused attention core: one workgroup = (16 queries) x (1 head) x (1 bm group).
// S[16x512] lives in LDS; masked softmax; ctx = P @ V with 4-wave split-K.
// Block: 128 threads = 4 waves. Wave w owns key columns [128w, 128w+128).
// ---------------------------------------------------------------------------
__global__ void __launch_bounds__(128) attn_kernel(const float* __restrict__ Qp,
                                                   const float* __restrict__ Kp,
                                                   const float* __restrict__ Vp,
                                                   const float* __restrict__ mask,
                                                   float* __restrict__ ctx) {
  __shared__ __align__(16) float S[16][LK];     // 32 KB score slab
  __shared__ float red[16][8];                  // softmax partials
  __shared__ float ctxp[4][16][16];             // per-wave ctx partials

  const int lane = threadIdx.x & 31;
  const int wave = threadIdx.x >> 5;
  const int lo = lane & 15, hi = lane >> 4;
  const int qb = blockIdx.x * 16;
  const int h = blockIdx.y;
  const int bm = blockIdx.z;
  const int batch = bm / Mm;

  // --- phase 1: S = (Q*scale) @ K^T + ext_mask -----------------------------
  const float* qrow = Qp + (size_t)(bm * AQ + qb + lo) * Dd + h * HD;
  v2f aq[4];
#pragma unroll
  for (int kk = 0; kk < 4; ++kk) {
    v2f t = *(const v2f*)(qrow + 4 * kk + 2 * hi);
    aq[kk].x = t.x * 0.25f;   // 1/sqrt(HD=16)
    aq[kk].y = t.y * 0.25f;
  }
  for (int ct = 0; ct < 8; ++ct) {
    const int cb = wave * 128 + ct * 16;
    const float* krow = Kp + (size_t)(bm * LK + cb + lo) * Dd + h * HD;
    v8f acc = {0.f, 0.f, 0.f, 0.f, 0.f, 0.f, 0.f, 0.f};
#pragma unroll
    for (int kk = 0; kk < 4; ++kk) {
      v2f b = *(const v2f*)(krow + 4 * kk + 2 * hi);
      acc = wmma4(aq[kk], b, acc);
    }
#pragma unroll
    for (int r = 0; r < 8; ++r) {
      const int m = r + 8 * hi;
      const int l = cb + lo;
      const float mv = mask[((size_t)batch * AQ + qb + m) * LK + l];
      S[m][l] = acc[r] + (1.0f - mv) * -10000.0f;
    }
  }
  __syncthreads();

  // --- phase 2: row softmax over 512 (8 threads per row) -------------------
  const int row = threadIdx.x >> 3;  // 0..15
  const int sub = threadIdx.x & 7;   // 0..7
  float lmax = -3.0e38f;
  for (int c = sub; c < LK; c += 8) lmax = fmaxf(lmax, S[row][c]);
  red[row][sub] = lmax;
  __syncthreads();
  float rmax = red[row][0];
#pragma unroll
  for (int i = 1; i < 8; ++i) rmax = fmaxf(rmax, red[row][i]);
  __syncthreads();
  float lsum = 0.0f;
  for (int c = sub; c < LK; c += 8) {
    const float e = expf(S[row][c] - rmax);
    S[row][c] = e;
    lsum += e;
  }
  red[row][sub] = lsum;
  __syncthreads();
  float rsum = 0.0f;
#pragma unroll
  for (int i = 0; i < 8; ++i) rsum += red[row][i];
  const float inv = 1.0f / rsum;
  for (int c = sub; c < LK; c += 8) S[row][c] *= inv;
  __syncthreads();

  // --- phase 3: ctx = P @ V, split-K across 4 waves ------------------------
  v8f acc = {0.f, 0.f, 0.f, 0.f, 0.f, 0.f, 0.f, 0.f};
  const int base = wave * 128;
  for (int kk = 0; kk < 32; ++kk) {
    const int k0 = base + 4 * kk + 2 * hi;
    v2f a = *(const v2f*)(&S[lo][k0]);
    v2f b;
    b.x = Vp[(size_t)(bm * LK + k0) * Dd + h * HD + lo];
    b.y = Vp[(size_t)(bm * LK + k0 + 1) * Dd + h * HD + lo];
    acc = wmma4(a, b, acc);
  }
#pragma unroll
  for (int r = 0; r < 8; ++r) ctxp[wave][r + 8 * hi][lo] = acc[r];
  __syncthreads();

  for (int idx = threadIdx.x; idx < 256; idx += 128) {
    const int m = idx >> 4, n = idx & 15;
    const float s = ctxp[0][m][n] + ctxp[1][m][n] + ctxp[2][m][n] + ctxp[3][m][n];
    ctx[(size_t)(bm * AQ + qb + m) * Dd + h * HD + n] = s;
  }
}

// ---------------------------------------------------------------------------
// y = LayerNorm(a [+ resid]) * g + beta  [, ReLU]   (rows of 128, 1 wave/row)
// ---------------------------------------------------------------------------
__global__ void __launch_bounds__(256) ln_kernel(const float* __restrict__ a,
                                                 const float* __restrict__ resid,
                                                 const float* __restrict__ g,
                                                 const float* __restrict__ beta,
                                                 float* __restrict__ y,
                                                 int relu) {
  const int lane = threadIdx.x & 31;
  const int wave = threadIdx.x >> 5;
  const size_t row = (size_t)blockIdx.x * 8 + wave;
  const int c = lane * 4;

  float4 v = *(const float4*)(a + row * Dd + c);
  if (resid) {
    float4 rr = *(const float4*)(resid + row * Dd + c);
    v.x += rr.x; v.y += rr.y; v.z += rr.z; v.w += rr.w;
  }
  float s = v.x + v.y + v.z + v.w;
  for (int off = 16; off; off >>= 1) s += __shfl_xor(s, off, 32);
  const float mean = s * (1.0f / Dd);
  const float dx = v.x - mean, dy = v.y - mean, dz = v.z - mean, dw = v.w - mean;
  float vs = dx * dx + dy * dy + dz * dz + dw * dw;
  for (int off = 16; off; off >>= 1) vs += __shfl_xor(vs, off, 32);
  const float ir = rsqrtf(vs * (1.0f / Dd) + 1e-5f);

  const float4 gg = *(const float4*)(g + c);
  const float4 bb = *(const float4*)(beta + c);
  float4 o;
  o.x = dx * ir * gg.x + bb.x;
  o.y = dy * ir * gg.y + bb.y;
  o.z = dz * ir * gg.z + bb.z;
  o.w = dw * ir * gg.w + bb.w;
  if (relu) {
    o.x = fmaxf(o.x, 0.f); o.y = fmaxf(o.y, 0.f);
    o.z = fmaxf(o.z, 0.f); o.w = fmaxf(o.w, 0.f);
  }
  *(float4*)(y + row * Dd + c) = o;
}

// ---------------------------------------------------------------------------
extern "C" void kernel_launch(void* const* d_in, const int* in_sizes, int n_in,
                              void* d_out, int out_size, void* d_ws, size_t ws_size,
                              hipStream_t stream) {
  (void)in_sizes; (void)n_in; (void)out_size; (void)ws_size;
  const float* query  = (const float*)d_in[0];
  const float* kv     = (const float*)d_in[1];
  const float* amask  = (const float*)d_in[2];
  const float* Wq     = (const float*)d_in[3];
  const float* bq     = (const float*)d_in[4];
  const float* Wk     = (const float*)d_in[5];
  const float* Wv     = (const float*)d_in[6];
  const float* bv     = (const float*)d_in[7];
  const float* Wo     = (const float*)d_in[8];
  const float* bo     = (const float*)d_in[9];
  const float* W1     = (const float*)d_in[10];
  const float* b1     = (const float*)d_in[11];
  const float* mg     = (const float*)d_in[12];
  const float* mb     = (const float*)d_in[13];
  const float* W2     = (const float*)d_in[14];
  const float* b2     = (const float*)d_in[15];
  const float* ln1g   = (const float*)d_in[16];
  const float* ln1b   = (const float*)d_in[17];
  const float* ln2g   = (const float*)d_in[18];
  const float* ln2b   = (const float*)d_in[19];

  const size_t nQ  = (size_t)Bb * Mm * AQ * Dd;   //  3,145,728 floats
  const size_t nKV = (size_t)Bb * Mm * LK * Dd;   // 12,582,912 floats
  float* ws  = (float*)d_ws;
  float* Qp  = ws;              // [nQ]
  float* Kp  = Qp + nQ;         // [nKV]
  float* Vp  = Kp + nKV;        // [nKV]
  float* ctx = Vp + nKV;        // [nQ]   peak ws = nQ*2 + nKV*2 = ~120 MB
  // post-attention reuse (K/V projections dead after attn_kernel):
  float* attn_out = Kp;         // [nQ]
  float* x1       = Kp + nQ;    // [nQ]
  float* t2       = Vp;         // [nQ]
  float* ffn      = Vp + nQ;    // [nQ]

  const int Rq  = Bb * Mm * AQ;  // 24576 rows
  const int Rkv = Bb * Mm * LK;  // 98304 rows

  proj128<<<Rq / 16, 256, 0, stream>>>(query, Wq, bq, Qp);
  proj128<<<Rkv / 16, 256, 0, stream>>>(kv, Wk, nullptr, Kp);   // no key bias
  proj128<<<Rkv / 16, 256, 0, stream>>>(kv, Wv, bv, Vp);

  attn_kernel<<<dim3(AQ / 16, Hh, Bb * Mm), 128, 0, stream>>>(Qp, Kp, Vp, amask, ctx);

  proj128<<<Rq / 16, 256, 0, stream>>>(ctx, Wo, bo, attn_out);
  ln_kernel<<<Rq / 8, 256, 0, stream>>>(attn_out, query, ln1g, ln1b, x1, 0);

  proj128<<<Rq / 16, 256, 0, stream>>>(x1, W1, b1, t2);
  ln_kernel<<<Rq / 8, 256, 0, stream>>>(t2, nullptr, mg, mb, t2, 1);   // in-place LN+ReLU
  proj128<<<Rq / 16, 256, 0, stream>>>(t2, W2, b2, ffn);
  ln_kernel<<<Rq / 8, 256, 0, stream>>>(ffn, x1, ln2g, ln2b, (float*)d_out, 0);
}